// HIMMOEActorCritic_27625229648261
// MI455X (gfx1250) — compile-verified
//
#include <hip/hip_runtime.h>

#define B_TOTAL   65536
#define OBS_DIM   225
#define ONE_DIM   45
#define AIN_DIM   64
#define CIN_DIM   238
#define NEXP      5
#define HDIM      256
#define TE_DIM    64
#define ADIM      12

#define WAVES         2
#define ROWS_PER_WAVE 16
#define MAIN_STRIDE   260   // (4*row + col) mod 64 banks -> conflict-free A loads
#define AIN_STRIDE    68

typedef __attribute__((ext_vector_type(2))) float v2f;
typedef __attribute__((ext_vector_type(8))) float v8f;

__device__ __forceinline__ v8f wmma4(v2f a, v2f b, v8f c) {
  // D = A(16x4 f32) * B(4x16 f32) + C(16x16 f32)
  return __builtin_amdgcn_wmma_f32_16x16x4_f32(
      /*neg_a=*/false, a, /*neg_b=*/false, b,
      /*c_mod=*/(short)0, c, /*reuse_a=*/false, /*reuse_b=*/false);
}

__device__ __forceinline__ float elu1(float x) {
  return x > 0.f ? x : (__expf(x) - 1.f);
}

// GEMM: A tile in LDS ([16 x ceil4(KMAX)], row-major, strideA, zero-padded past
// KMAX), B streamed from global row-major [KMAX x ldn]. Template params make all
// guards compile-time; the N guard (NMAX < NT*16) and the K tail guard are
// branch-free: clamp index (always in-bounds) + multiply by 0/1 mask.
// A-frag layout per ISA 7.12.2 (f32 16x4): lanes 0-15 hold K=k0,k0+1;
// lanes 16-31 hold K=k0+2,k0+3 (same M = lane&15).
template<int NT, int NMAX, int KMAX>
__device__ __forceinline__ void gemm_tile(const float* tileA, int strideA,
                                          const float* __restrict__ W, int ldn,
                                          int lane, v8f* c) {
  constexpr int  KFULL    = KMAX / 4;
  constexpr bool HAS_TAIL = (KMAX % 4) != 0;
  constexpr bool NGUARD   = (NMAX < NT * 16);

  v8f z = {0.f,0.f,0.f,0.f,0.f,0.f,0.f,0.f};
  #pragma unroll
  for (int nt = 0; nt < NT; ++nt) c[nt] = z;

  const int row   = lane & 15;
  const int khalf = (lane >> 4) << 1;
  const int n0    = lane & 15;

  for (int ks = 0; ks < KFULL; ++ks) {
    const int k0 = ks * 4;
    const int kk = k0 + khalf;
    v2f a = *(const v2f*)(tileA + row * strideA + kk);   // ds_load_b64
    #pragma unroll
    for (int nt = 0; nt < NT; ++nt) {
      const int n = nt * 16 + n0;
      v2f b;
      if (!NGUARD) {
        b.x = W[(size_t)kk * ldn + n];
        b.y = W[(size_t)(kk + 1) * ldn + n];
      } else {
        const int   nc = (n < NMAX) ? n : (NMAX - 1);    // always in-bounds
        const float nm = (n < NMAX) ? 1.f : 0.f;
        b.x = W[(size_t)kk * ldn + nc] * nm;
        b.y = W[(size_t)(kk + 1) * ldn + nc] * nm;
      }
      c[nt] = wmma4(a, b, c[nt]);
    }
  }

  if (HAS_TAIL) {
    const int k0 = KFULL * 4;
    const int kk = k0 + khalf;
    v2f a = *(const v2f*)(tileA + row * strideA + kk);   // LDS is zero-padded
    #pragma unroll
    for (int nt = 0; nt < NT; ++nt) {
      const int   n  = nt * 16 + n0;
      const int   nc = NGUARD ? ((n < NMAX) ? n : (NMAX - 1)) : n;
      const float nm = NGUARD ? ((n < NMAX) ? 1.f : 0.f) : 1.f;
      const int   kc0 = (kk     < KMAX) ? kk       : (KMAX - 1);
      const int   kc1 = (kk + 1 < KMAX) ? (kk + 1) : (KMAX - 1);
      const float m0  = (kk     < KMAX) ? nm : 0.f;
      const float m1  = (kk + 1 < KMAX) ? nm : 0.f;
      v2f b;
      b.x = W[(size_t)kc0 * ldn + nc] * m0;
      b.y = W[(size_t)kc1 * ldn + nc] * m1;
      c[nt] = wmma4(a, b, c[nt]);
    }
  }
}

// Write C tiles (bias + optional ELU) back into an LDS A-tile for the next GEMM.
// C layout: VGPR j, lanes 0-15: M=j; lanes 16-31: M=8+j; N = nt*16 + (lane&15).
template<int NT, bool ACT>
__device__ __forceinline__ void store_c_lds(float* tile, int strideT,
                                            const v8f* c,
                                            const float* __restrict__ bias, int lane) {
  const int mhi = (lane >> 4) << 3;
  const int n0  = lane & 15;
  #pragma unroll
  for (int nt = 0; nt < NT; ++nt) {
    const float bv = bias[nt * 16 + n0];
    #pragma unroll
    for (int j = 0; j < 8; ++j) {
      float v = c[nt][j] + bv;
      if (ACT) v = elu1(v);
      tile[(mhi + j) * strideT + nt * 16 + n0] = v;
    }
  }
}

__global__ __launch_bounds__(32 * WAVES)
void moe_actor_critic_kernel(
    const float* __restrict__ obs,  const float* __restrict__ vel,
    const float* __restrict__ lat,  const float* __restrict__ cobs,
    const float* __restrict__ Wte,  const float* __restrict__ bte,
    const float* __restrict__ Wg1,  const float* __restrict__ bg1,
    const float* __restrict__ Wg2,  const float* __restrict__ bg2,
    const float* __restrict__ aW1,  const float* __restrict__ ab1,
    const float* __restrict__ aW2,  const float* __restrict__ ab2,
    const float* __restrict__ aW3,  const float* __restrict__ ab3,
    const float* __restrict__ cW1,  const float* __restrict__ cb1,
    const float* __restrict__ cW2,  const float* __restrict__ cb2,
    const float* __restrict__ cW3,  const float* __restrict__ cb3,
    float* __restrict__ out_actor, float* __restrict__ out_critic,
    float* __restrict__ out_gate)
{
  __shared__ float s_main[WAVES][ROWS_PER_WAVE * MAIN_STRIDE];
  __shared__ float s_ain [WAVES][ROWS_PER_WAVE * AIN_STRIDE];
  __shared__ float s_gate[WAVES][ROWS_PER_WAVE * 8];

  const int lane = threadIdx.x & 31;
  const int wv   = threadIdx.x >> 5;
  const int r0   = (blockIdx.x * WAVES + wv) * ROWS_PER_WAVE;

  float* mainT = s_main[wv];
  float* ainT  = s_ain[wv];
  float* gateT = s_gate[wv];

  const int n0  = lane & 15;
  const int mhi = (lane >> 4) << 3;

  // ---- stage actor input: concat(obs[:, -45:], vel, latent) -> [16 x 64]
  for (int i = lane; i < ROWS_PER_WAVE * AIN_DIM; i += 32) {
    const int row = i >> 6, col = i & 63;
    float v;
    if (col < ONE_DIM)          v = obs[(size_t)(r0 + row) * OBS_DIM + (OBS_DIM - ONE_DIM) + col];
    else if (col < ONE_DIM + 3) v = vel[(size_t)(r0 + row) * 3 + (col - ONE_DIM)];
    else                        v = lat[(size_t)(r0 + row) * 16 + (col - ONE_DIM - 3)];
    ainT[row * AIN_STRIDE + col] = v;
  }

  // ---- stage observations [16 x 225] (zero-padded to K=228)
  for (int i = lane; i < ROWS_PER_WAVE * 228; i += 32) {
    const int row = i / 228, col = i % 228;
    mainT[row * MAIN_STRIDE + col] =
        (col < OBS_DIM) ? obs[(size_t)(r0 + row) * OBS_DIM + col] : 0.f;
  }

  v8f c16[16];

  // ---- te = ELU(obs @ Wte + bte)          [16 x 64]
  gemm_tile<4, TE_DIM, OBS_DIM>(mainT, MAIN_STRIDE, Wte, TE_DIM, lane, c16);
  store_c_lds<4, true>(mainT, MAIN_STRIDE, c16, bte, lane);

  // ---- g1 = ELU(te @ Wg1 + bg1)           [16 x 256]
  gemm_tile<16, HDIM, TE_DIM>(mainT, MAIN_STRIDE, Wg1, HDIM, lane, c16);
  store_c_lds<16, true>(mainT, MAIN_STRIDE, c16, bg1, lane);

  // ---- logits = g1 @ Wg2 + bg2            [16 x 5], then softmax
  gemm_tile<1, NEXP, HDIM>(mainT, MAIN_STRIDE, Wg2, NEXP, lane, c16);
  if (n0 < NEXP) {
    #pragma unroll
    for (int j = 0; j < 8; ++j)
      gateT[(mhi + j) * 8 + n0] = c16[0][j] + bg2[n0];
  }
  if (lane < 16) {  // one row per lane
    float v[NEXP], m = -3.4e38f;
    #pragma unroll
    for (int e = 0; e < NEXP; ++e) { v[e] = gateT[lane * 8 + e]; m = fmaxf(m, v[e]); }
    float s = 0.f;
    #pragma unroll
    for (int e = 0; e < NEXP; ++e) { v[e] = __expf(v[e] - m); s += v[e]; }
    const float inv = 1.f / s;
    #pragma unroll
    for (int e = 0; e < NEXP; ++e) {
      const float g = v[e] * inv;
      gateT[lane * 8 + e] = g;
      out_gate[(size_t)(r0 + lane) * NEXP + e] = g;
    }
  }

  // ---- actor experts: AIN(64)->H->H->A(12), gate-weighted sum
  v8f accA = {0.f,0.f,0.f,0.f,0.f,0.f,0.f,0.f};
  for (int e = 0; e < NEXP; ++e) {
    gemm_tile<16, HDIM, AIN_DIM>(ainT, AIN_STRIDE, aW1 + (size_t)e * AIN_DIM * HDIM, HDIM, lane, c16);
    store_c_lds<16, true>(mainT, MAIN_STRIDE, c16, ab1 + e * HDIM, lane);
    gemm_tile<16, HDIM, HDIM>(mainT, MAIN_STRIDE, aW2 + (size_t)e * HDIM * HDIM, HDIM, lane, c16);
    store_c_lds<16, true>(mainT, MAIN_STRIDE, c16, ab2 + e * HDIM, lane);
    gemm_tile<1, ADIM, HDIM>(mainT, MAIN_STRIDE, aW3 + (size_t)e * HDIM * ADIM, ADIM, lane, c16);
    const float bv = (n0 < ADIM) ? ab3[e * ADIM + n0] : 0.f;
    #pragma unroll
    for (int j = 0; j < 8; ++j) {
      const float g = gateT[(mhi + j) * 8 + e];
      accA[j] += g * (c16[0][j] + bv);
    }
  }

  // ---- critic experts: CIN(238)->H->H->1, gate-weighted sum
  v8f accC = {0.f,0.f,0.f,0.f,0.f,0.f,0.f,0.f};
  for (int e = 0; e < NEXP; ++e) {
    for (int i = lane; i < ROWS_PER_WAVE * 240; i += 32) {     // re-stage (L2-hot)
      const int row = i / 240, col = i % 240;
      mainT[row * MAIN_STRIDE + col] =
          (col < CIN_DIM) ? cobs[(size_t)(r0 + row) * CIN_DIM + col] : 0.f;
    }
    gemm_tile<16, HDIM, CIN_DIM>(mainT, MAIN_STRIDE, cW1 + (size_t)e * CIN_DIM * HDIM, HDIM, lane, c16);
    store_c_lds<16, true>(mainT, MAIN_STRIDE, c16, cb1 + e * HDIM, lane);
    gemm_tile<16, HDIM, HDIM>(mainT, MAIN_STRIDE, cW2 + (size_t)e * HDIM * HDIM, HDIM, lane, c16);
    store_c_lds<16, true>(mainT, MAIN_STRIDE, c16, cb2 + e * HDIM, lane);
    gemm_tile<1, 1, HDIM>(mainT, MAIN_STRIDE, cW3 + (size_t)e * HDIM, 1, lane, c16);
    const float bv = (n0 == 0) ? cb3[e] : 0.f;
    #pragma unroll
    for (int j = 0; j < 8; ++j) {
      const float g = gateT[(mhi + j) * 8 + e];
      accC[j] += g * (c16[0][j] + bv);
    }
  }

  // ---- outputs
  #pragma unroll
  for (int j = 0; j < 8; ++j) {
    const int r = r0 + mhi + j;
    if (n0 < ADIM) out_actor[(size_t)r * ADIM + n0] = accA[j];
    if (n0 == 0)   out_critic[r] = accC[j];
  }
}

extern "C" void kernel_launch(void* const* d_in, const int* in_sizes, int n_in,
                              void* d_out, int out_size, void* d_ws, size_t ws_size,
                              hipStream_t stream) {
  (void)in_sizes; (void)n_in; (void)d_ws; (void)ws_size; (void)out_size;
  const float* obs = (const float*)d_in[0];
  const float* vel = (const float*)d_in[1];
  const float* lat = (const float*)d_in[2];
  const float* cob = (const float*)d_in[3];
  const float* Wte = (const float*)d_in[4];
  const float* bte = (const float*)d_in[5];
  const float* Wg1 = (const float*)d_in[6];
  const float* bg1 = (const float*)d_in[7];
  const float* Wg2 = (const float*)d_in[8];
  const float* bg2 = (const float*)d_in[9];
  const float* aW1 = (const float*)d_in[10];
  const float* ab1 = (const float*)d_in[11];
  const float* aW2 = (const float*)d_in[12];
  const float* ab2 = (const float*)d_in[13];
  const float* aW3 = (const float*)d_in[14];
  const float* ab3 = (const float*)d_in[15];
  const float* cW1 = (const float*)d_in[16];
  const float* cb1 = (const float*)d_in[17];
  const float* cW2 = (const float*)d_in[18];
  const float* cb2 = (const float*)d_in[19];
  const float* cW3 = (const float*)d_in[20];
  const float* cb3 = (const float*)d_in[21];

  float* out        = (float*)d_out;
  float* out_actor  = out;                               // [B,12]
  float* out_critic = out + (size_t)B_TOTAL * ADIM;      // [B,1]
  float* out_gate   = out + (size_t)B_TOTAL * (ADIM + 1);// [B,5]

  const int rows_per_block = WAVES * ROWS_PER_WAVE;      // 32
  dim3 grid(B_TOTAL / rows_per_block);                   // 2048
  dim3 block(32 * WAVES);                                // 64

  moe_actor_critic_kernel<<<grid, block, 0, stream>>>(
      obs, vel, lat, cob, Wte, bte, Wg1, bg1, Wg2, bg2,
      aW1, ab1, aW2, ab2, aW3, ab3, cW1, cb1, cW2, cb2, cW3, cb3,
      out_actor, out_critic, out_gate);
}